// GConv_64750926955121
// MI455X (gfx1250) — compile-verified
//
#include <hip/hip_runtime.h>

typedef __attribute__((ext_vector_type(2))) float v2f;
typedef __attribute__((ext_vector_type(8))) float v8f;

#define HDIM 128

// ---------------- utility kernels ----------------

__global__ void k_fill(float* __restrict__ p, float v, int n) {
    int i = blockIdx.x * blockDim.x + threadIdx.x;
    if (i < n) p[i] = v;
}

__global__ void k_deg_count(const int* __restrict__ dst, float* __restrict__ deg, int E) {
    int i = blockIdx.x * blockDim.x + threadIdx.x;
    if (i < E) atomicAdd(&deg[dst[i]], 1.0f);
}

__global__ void k_rsqrt_inplace(float* __restrict__ p, int n) {
    int i = blockIdx.x * blockDim.x + threadIdx.x;
    if (i < n) p[i] = rsqrtf(p[i]);
}

// ---------------- WMMA GEMM:  C[M,H] = A[M,K] * B[K,H]  (K = H = 128, f32) ----------------
// One wave computes one 16x16 tile via 32 chained V_WMMA_F32_16X16X4_F32 ops.
// Block = 256 threads = 8 waves = one 16-row stripe across all 8 column tiles.
__global__ __launch_bounds__(256) void gemm_wmma_f32(
    const float* __restrict__ A, const float* __restrict__ B,
    float* __restrict__ C, int M) {
    const int K = HDIM;
    const int wave = threadIdx.x >> 5;
    const int lane = threadIdx.x & 31;

    const int m0 = blockIdx.x * 16;        // row tile base
    if (m0 >= M) return;                   // block-uniform guard (EXEC stays all-ones)
    const int n0 = wave * 16;              // col tile base (H/16 = 8 waves)

    const int lmod = lane & 15;
    const int kb   = (lane >> 4) << 1;     // 0 for lanes 0-15, 2 for lanes 16-31

    const float* __restrict__ arow = A + (size_t)(m0 + lmod) * K;
    const float* __restrict__ bcol = B + (n0 + lmod);

    v8f acc = {};
#pragma unroll
    for (int k = 0; k < K; k += 4) {
        v2f a, b;
        a.x = arow[k + kb];
        a.y = arow[k + kb + 1];
        b.x = bcol[(size_t)(k + kb) * HDIM];
        b.y = bcol[(size_t)(k + kb + 1) * HDIM];
        acc = __builtin_amdgcn_wmma_f32_16x16x4_f32(
            false, a, false, b, (short)0, acc, false, false);
    }

    // D layout: VGPR r -> row m0 + r + (lane>=16 ? 8 : 0), col n0 + (lane&15)
    const int rbase = m0 + ((lane >> 4) << 3);
    float* __restrict__ cptr = C + (size_t)rbase * HDIM + n0 + lmod;
#pragma unroll
    for (int r = 0; r < 8; ++r) {
        cptr[(size_t)r * HDIM] = acc[r];
    }
}

// ---------------- edge scatter:  agg[dst] += zw[src] * dinv[src]*dinv[dst] ----------------
// One wave per edge (incl. N virtual self-loop edges); each lane moves a float4.
__global__ __launch_bounds__(256) void k_scatter(
    const float* __restrict__ zw, const int* __restrict__ src, const int* __restrict__ dst,
    const float* __restrict__ dinv, float* __restrict__ agg, int E, int N) {
    const int wavesPerBlock = blockDim.x >> 5;
    const int e = blockIdx.x * wavesPerBlock + (threadIdx.x >> 5);
    if (e >= E + N) return;                 // wave-uniform guard
    const int lane = threadIdx.x & 31;

    int s, d;
    if (e < E) { s = src[e]; d = dst[e]; }
    else       { s = e - E;  d = s;      }  // self-loop
    const float norm = dinv[s] * dinv[d];

    const float4 v = *(const float4*)(zw + (size_t)s * HDIM + lane * 4);
    float* __restrict__ ap = agg + (size_t)d * HDIM + lane * 4;
    atomicAdd(ap + 0, v.x * norm);
    atomicAdd(ap + 1, v.y * norm);
    atomicAdd(ap + 2, v.z * norm);
    atomicAdd(ap + 3, v.w * norm);
}

// ---------------- bias + PReLU + write z + pool into g ----------------
__global__ void k_post(const float* __restrict__ agg, const float* __restrict__ bias,
                       const float* __restrict__ alpha, const int* __restrict__ batch,
                       float* __restrict__ zout, float* __restrict__ gpool,
                       int goffset, int N) {
    int i = blockIdx.x * blockDim.x + threadIdx.x;
    if (i >= N * HDIM) return;
    const int h = i & (HDIM - 1);
    const int n = i >> 7;
    float z = agg[i] + bias[h];
    z = (z > 0.0f) ? z : alpha[h] * z;
    zout[i] = z;
    atomicAdd(&gpool[(size_t)batch[n] * (2 * HDIM) + goffset + h], z);
}

// ---------------- launch ----------------

extern "C" void kernel_launch(void* const* d_in, const int* in_sizes, int n_in,
                              void* d_out, int out_size, void* d_ws, size_t ws_size,
                              hipStream_t stream) {
    const float* x     = (const float*)d_in[0];
    const int*   ei    = (const int*)  d_in[1];
    const int*   batch = (const int*)  d_in[2];
    const float* W0    = (const float*)d_in[3];
    const float* b0    = (const float*)d_in[4];
    const float* al0   = (const float*)d_in[5];
    const float* W1    = (const float*)d_in[6];
    const float* b1    = (const float*)d_in[7];
    const float* al1   = (const float*)d_in[8];

    const int N = in_sizes[2];
    const int E = in_sizes[1] / 2;
    const int* srcp = ei;
    const int* dstp = ei + E;

    float* ws   = (float*)d_ws;
    float* dinv = ws;                         // [N]
    float* zw   = dinv + N;                   // [N*H]
    float* agg  = zw  + (size_t)N * HDIM;     // [N*H]
    float* z1   = agg + (size_t)N * HDIM;     // [N*H]

    float* z2out = (float*)d_out;             // [N*H]
    float* gpool = z2out + (size_t)N * HDIM;  // [G*2H]
    const int gsize = out_size - N * HDIM;

    const int T = 256;
    const int NH = N * HDIM;
    const int nEdges = E + N;
    const int scatterBlocks = (nEdges + 7) / 8;   // 8 waves per 256-thread block
    const int gemmBlocks = (N + 15) / 16;

    // degrees -> dinv (self-loop counted by init value 1.0)
    k_fill<<<(N + T - 1) / T, T, 0, stream>>>(dinv, 1.0f, N);
    k_deg_count<<<(E + T - 1) / T, T, 0, stream>>>(dstp, dinv, E);
    k_rsqrt_inplace<<<(N + T - 1) / T, T, 0, stream>>>(dinv, N);

    // zero pooled-output region of d_out
    k_fill<<<(gsize + T - 1) / T, T, 0, stream>>>(gpool, 0.0f, gsize);

    // ---- layer 0 ----
    gemm_wmma_f32<<<gemmBlocks, 256, 0, stream>>>(x, W0, zw, N);
    k_fill<<<(NH + T - 1) / T, T, 0, stream>>>(agg, 0.0f, NH);
    k_scatter<<<scatterBlocks, 256, 0, stream>>>(zw, srcp, dstp, dinv, agg, E, N);
    k_post<<<(NH + T - 1) / T, T, 0, stream>>>(agg, b0, al0, batch, z1, gpool, 0, N);

    // ---- layer 1 ----
    gemm_wmma_f32<<<gemmBlocks, 256, 0, stream>>>(z1, W1, zw, N);
    k_fill<<<(NH + T - 1) / T, T, 0, stream>>>(agg, 0.0f, NH);
    k_scatter<<<scatterBlocks, 256, 0, stream>>>(zw, srcp, dstp, dinv, agg, E, N);
    k_post<<<(NH + T - 1) / T, T, 0, stream>>>(agg, b1, al1, batch, z2out, gpool, HDIM, N);
}